// Attention_4767413699310
// MI455X (gfx1250) — compile-verified
//
#include <hip/hip_runtime.h>
#include <math.h>

// ---------------------------------------------------------------------------
// Attention (qkv 1x1 conv -> token-axis L2 norm -> scaled dot-prod attention
// with online softmax -> out 1x1 conv), fp32 end-to-end using
// V_WMMA_F32_16X16X4_F32 on gfx1250 (wave32), with TDM (tensor_load_to_lds)
// double-buffered K/V staging in the flash-attention kernel.
// ---------------------------------------------------------------------------

typedef __attribute__((ext_vector_type(2))) float v2f;
typedef __attribute__((ext_vector_type(8))) float v8f;
typedef __attribute__((ext_vector_type(4))) unsigned int u32x4;
typedef __attribute__((ext_vector_type(8))) int i32x8;

#define B_    4
#define S_    4096          // H*W tokens per batch
#define C_    128
#define NH    4
#define DH    32
#define DIM   128           // NH*DH
#define F3    384           // 3*DIM
#define SCALE_ 10.0f

static __device__ __forceinline__ v8f wmma_f32(v2f a, v2f b, v8f c) {
  // D(16x16,f32) = A(16x4,f32) * B(4x16,f32) + C
  return __builtin_amdgcn_wmma_f32_16x16x4_f32(
      /*neg_a=*/false, a, /*neg_b=*/false, b,
      /*c_mod=*/(short)0, c, /*reuse_a=*/false, /*reuse_b=*/false);
}

// ---------------------------------------------------------------------------
// TDM 2D tile load: D# group0/group1 packed per CDNA5 ISA §8.3/8.4.
//   group0: count=1 (valid), lds_addr, global_addr[56:0], type=2 ("image")
//   group1: data_size=4B, tensor_dim0/1, tile_dim0/1 (tile_dim2=0 -> 2D),
//           tensor_dim0_stride (elements). workgroup_mask=0 (no cluster).
// VADDR2/VADDR3 omitted (NULL) -> 2-group descriptor, tensors up to 2D.
// Tracked with TENSORcnt.
// ---------------------------------------------------------------------------
static __device__ __forceinline__ void tdm_load_2d(
    unsigned lds_off, const void* gaddr,
    unsigned tdim0, unsigned tdim1,
    unsigned tiledim0, unsigned tiledim1,
    unsigned long long stride0_elems)
{
  const unsigned long long ga = (unsigned long long)(uintptr_t)gaddr;
  u32x4 g0;
  g0.x = 1u;                                                   // count=1, user desc
  g0.y = lds_off;                                              // LDS byte address
  g0.z = (unsigned)ga;                                         // global_addr[31:0]
  g0.w = (unsigned)((ga >> 32) & 0x01FFFFFFu) | 0x80000000u;   // [56:32] | type=2
  i32x8 g1;
  g1[0] = (int)0x00020000u;                                    // data_size=4B
  g1[1] = (int)((tdim0 & 0xFFFFu) << 16);                      // tensor_dim0 lo
  g1[2] = (int)(((tdim0 >> 16) & 0xFFFFu) | ((tdim1 & 0xFFFFu) << 16));
  g1[3] = (int)(((tdim1 >> 16) & 0xFFFFu) | ((tiledim0 & 0xFFFFu) << 16));
  g1[4] = (int)(tiledim1 & 0xFFFFu);                           // tile_dim1, tile_dim2=0
  g1[5] = (int)(unsigned)(stride0_elems & 0xFFFFFFFFull);      // dim0 stride lo
  g1[6] = (int)(unsigned)((stride0_elems >> 32) & 0xFFFFull);  // dim0 stride hi
  g1[7] = 0;                                                   // dim1 stride (unused)
  asm volatile("tensor_load_to_lds %0, %1" :: "s"(g0), "s"(g1) : "memory");
}

// ---------------------------------------------------------------------------
// Kernel 1: qkv = x @ w_qkv  (16384x128 @ 128x384)
// q,k written transposed (d-major: [b][h][d][s]); v written [b][h][s][d].
// One wave owns a 16x16 output tile; 32 K-steps of WMMA f32 16x16x4.
// ---------------------------------------------------------------------------
__global__ __launch_bounds__(128) void qkv_gemm(
    const float* __restrict__ x, const float* __restrict__ w,
    float* __restrict__ qT, float* __restrict__ kT, float* __restrict__ v)
{
  const int lane = threadIdx.x & 31;
  const int wave = threadIdx.x >> 5;
  const int tile = blockIdx.x * 4 + wave;       // 24576 tiles total
  const int mt   = tile / (F3 / 16);            // 0..1023 row tiles
  const int nt   = tile % (F3 / 16);            // 0..23 col tiles
  const int m    = lane & 15;                   // A row / B col within tile
  const int kb   = (lane >> 4) << 1;            // K sub-offset per lane half
  const int row0 = mt * 16;
  const int f0   = nt * 16;

  v8f acc = {};
  const float* xr = x + (size_t)(row0 + m) * C_;
#pragma unroll
  for (int k0 = 0; k0 < C_; k0 += 4) {
    v2f a;  a.x = xr[k0 + kb];               a.y = xr[k0 + kb + 1];
    v2f bb; bb.x = w[(k0 + kb) * F3 + f0 + m]; bb.y = w[(k0 + kb + 1) * F3 + f0 + m];
    acc = wmma_f32(a, bb, acc);
  }

  const int half = lane >> 4;
#pragma unroll
  for (int r = 0; r < 8; ++r) {
    const int M    = r + 8 * half;
    const int row  = row0 + M;                  // global token index (b*S + s)
    const int f    = f0 + m;                    // output feature 0..383
    const int part = f / DIM;                   // 0=q 1=k 2=v (uniform per tile)
    const int c    = f % DIM;
    const int h    = c / DH;
    const int d    = c % DH;
    const int b    = row / S_;
    const int s    = row % S_;
    const float val = acc[r];
    if (part == 0)
      qT[((size_t)(b * NH + h) * DH + d) * S_ + s] = val;
    else if (part == 1)
      kT[((size_t)(b * NH + h) * DH + d) * S_ + s] = val;
    else
      v[((size_t)(b * NH + h) * S_ + s) * DH + d] = val;
  }
}

// ---------------------------------------------------------------------------
// Kernel 2: token-axis L2 norm. One block per (part,b,h,d) column of 4096
// contiguous floats; reduce sum-of-squares, then scale in place.
// q gets SCALE/denom (folds the *10), k gets 1/denom.
// ---------------------------------------------------------------------------
__global__ __launch_bounds__(256) void l2norm_scale(
    float* __restrict__ qT, float* __restrict__ kT)
{
  __shared__ float red[256];
  const int col  = blockIdx.x;                  // 0..1023
  const int part = col >> 9;                    // 0=q, 1=k
  const int idx  = col & 511;                   // b*128 + h*32 + d
  float* base = (part ? kT : qT) + (size_t)idx * S_;

  float ss = 0.f;
  for (int i = threadIdx.x; i < S_; i += 256) { float t = base[i]; ss += t * t; }
  red[threadIdx.x] = ss;
  __syncthreads();
  for (int off = 128; off > 0; off >>= 1) {
    if ((int)threadIdx.x < off) red[threadIdx.x] += red[threadIdx.x + off];
    __syncthreads();
  }
  const float denom = fmaxf(sqrtf(red[0]), 1e-12f);
  const float scale = (part ? 1.0f : SCALE_) / denom;
  for (int i = threadIdx.x; i < S_; i += 256) base[i] *= scale;
}

// ---------------------------------------------------------------------------
// Kernel 3: flash attention. One block = 8 waves, all sharing one (b,h);
// each wave owns a 16-row Q tile. K/V tiles (32 keys) are staged into LDS by
// the Tensor Data Mover, double-buffered and overlapped with compute:
//   wave0: s_wait_tensorcnt 0 -> barrier -> issue DMA for next buffer ->
//   all waves compute QK^T / online softmax / PV from the current buffer.
// P is transposed through padded per-wave LDS into A-fragments for PV.
// ---------------------------------------------------------------------------
__global__ __launch_bounds__(256) void flash_attn(
    const float* __restrict__ qT, const float* __restrict__ kT,
    const float* __restrict__ v,  float* __restrict__ o)
{
  __shared__ float Klds[2][DH * 32];            // [buf][d*32 + j]   (K^T tile)
  __shared__ float Vlds[2][32 * DH];            // [buf][j*32 + d]   (V tile)
  __shared__ float Pshm[8][16 * 34];            // per-wave 16x32 P, stride 34

  const int lane = threadIdx.x & 31;
  const int wave = threadIdx.x >> 5;
  const int bh   = blockIdx.x >> 5;             // 0..15  (b*4+h), same for block
  const int qt   = (blockIdx.x & 31) * 8 + wave;// 0..255 Q tile
  const int b    = bh >> 2;
  const int h    = bh & 3;
  const int m    = lane & 15;
  const int kb2  = (lane >> 4) << 1;
  const int half = lane >> 4;

  const float* qb = qT + (size_t)bh * DH * S_;  // [d][s]
  const float* kp = kT + (size_t)bh * DH * S_;  // [d][s]
  const float* vb = v  + (size_t)bh * S_ * DH;  // [s][d]
  float* P = &Pshm[wave][0];

  // Preload Q A-fragments for all 8 K-steps of the 16x32 Q tile.
  v2f aq[8];
#pragma unroll
  for (int kk = 0; kk < 8; ++kk) {
    const int d0 = kk * 4 + kb2;
    aq[kk].x = qb[(size_t)d0 * S_ + qt * 16 + m];
    aq[kk].y = qb[(size_t)(d0 + 1) * S_ + qt * 16 + m];
  }

  v8f o0 = {}, o1 = {};
  float mrow[8], lrow[8];
#pragma unroll
  for (int r = 0; r < 8; ++r) { mrow[r] = -__builtin_huge_valf(); lrow[r] = 0.f; }

  // Prologue: kick off TDM for the first key tile (wave0 only; uniform branch).
  if (wave == 0) {
    tdm_load_2d((unsigned)(uintptr_t)&Klds[0][0], kp,
                /*tdim0=*/S_, /*tdim1=*/DH, /*tile0=*/32, /*tile1=*/DH,
                /*stride0=*/S_);
    tdm_load_2d((unsigned)(uintptr_t)&Vlds[0][0], vb,
                /*tdim0=*/32u * DH, /*tdim1=*/1, /*tile0=*/32u * DH, /*tile1=*/1,
                /*stride0=*/32u * DH);
  }

  for (int it = 0; it < S_ / 32; ++it) {
    const int buf = it & 1;
    const int j0  = it * 32;
    const float* Kb = &Klds[buf][0];
    const float* Vb = &Vlds[buf][0];

    if (wave == 0) __builtin_amdgcn_s_wait_tensorcnt(0);  // buf is resident
    __syncthreads();            // releases buf; all waves done reading buf^1
    if (wave == 0 && it + 1 < S_ / 32) {
      const int jn = j0 + 32;
      tdm_load_2d((unsigned)(uintptr_t)&Klds[buf ^ 1][0], kp + jn,
                  S_, DH, 32, DH, S_);
      tdm_load_2d((unsigned)(uintptr_t)&Vlds[buf ^ 1][0], vb + (size_t)jn * DH,
                  32u * DH, 1, 32u * DH, 1, 32u * DH);
    }

    // --- sim tile 16x32 = Q(16x32) @ K^T(32x32), K^T from LDS ---
    v8f s0 = {}, s1 = {};
#pragma unroll
    for (int kk = 0; kk < 8; ++kk) {
      const int d0 = kk * 4 + kb2;
      v2f bk;
      bk.x = Kb[d0 * 32 + m];
      bk.y = Kb[(d0 + 1) * 32 + m];
      s0 = wmma_f32(aq[kk], bk, s0);
      bk.x = Kb[d0 * 32 + 16 + m];
      bk.y = Kb[(d0 + 1) * 32 + 16 + m];
      s1 = wmma_f32(aq[kk], bk, s1);
    }

    // --- online softmax update; row i = r + 8*half lives in VGPR r of a half ---
#pragma unroll
    for (int r = 0; r < 8; ++r) {
      float mx = fmaxf(s0[r], s1[r]);
#pragma unroll
      for (int mk = 1; mk < 16; mk <<= 1) mx = fmaxf(mx, __shfl_xor(mx, mk, 32));
      const float mnew = fmaxf(mrow[r], mx);
      const float fac  = __expf(mrow[r] - mnew);
      const float p0   = __expf(s0[r] - mnew);
      const float p1   = __expf(s1[r] - mnew);
      float rs = p0 + p1;
#pragma unroll
      for (int mk = 1; mk < 16; mk <<= 1) rs += __shfl_xor(rs, mk, 32);
      lrow[r] = lrow[r] * fac + rs;
      mrow[r] = mnew;
      o0[r] *= fac;
      o1[r] *= fac;
      const int row = r + 8 * half;
      P[row * 34 + m]      = p0;                // D-frag layout -> LDS
      P[row * 34 + 16 + m] = p1;
    }

    // --- transpose P via LDS into A-fragments (8B-aligned, conflict-free) ---
    v2f ap[8];
#pragma unroll
    for (int kk = 0; kk < 8; ++kk)
      ap[kk] = *(const v2f*)&P[m * 34 + kk * 4 + kb2];

    // --- O(16x32) += P(16x32) @ V(32x32), V from LDS ---
#pragma unroll
    for (int kk = 0; kk < 8; ++kk) {
      const int jr = kk * 4 + kb2;
      v2f bv;
      bv.x = Vb[jr * 32 + m];
      bv.y = Vb[(jr + 1) * 32 + m];
      o0 = wmma_f32(ap[kk], bv, o0);
      bv.x = Vb[jr * 32 + 16 + m];
      bv.y = Vb[(jr + 1) * 32 + 16 + m];
      o1 = wmma_f32(ap[kk], bv, o1);
    }
  }

  // --- epilogue: normalize by row sum, write [b][s][h*32+d] ---
#pragma unroll
  for (int r = 0; r < 8; ++r) {
    const float inv = 1.0f / lrow[r];
    const int row = qt * 16 + r + 8 * half;
    float* op = o + ((size_t)(b * S_ + row)) * DIM + h * DH;
    op[m]      = o0[r] * inv;
    op[16 + m] = o1[r] * inv;
  }
}

// ---------------------------------------------------------------------------
// Kernel 4: y = o @ w_out + b_out  (16384x128 @ 128x128)
// ---------------------------------------------------------------------------
__global__ __launch_bounds__(128) void out_gemm(
    const float* __restrict__ o, const float* __restrict__ w,
    const float* __restrict__ bias, float* __restrict__ y)
{
  const int lane = threadIdx.x & 31;
  const int wave = threadIdx.x >> 5;
  const int tile = blockIdx.x * 4 + wave;       // 8192 tiles
  const int mt   = tile >> 3;
  const int nt   = tile & 7;
  const int m    = lane & 15;
  const int kb   = (lane >> 4) << 1;
  const int row0 = mt * 16;
  const int f0   = nt * 16;

  v8f acc = {};
  const float* orow = o + (size_t)(row0 + m) * DIM;
#pragma unroll
  for (int k0 = 0; k0 < DIM; k0 += 4) {
    v2f a;  a.x = orow[k0 + kb];               a.y = orow[k0 + kb + 1];
    v2f bb; bb.x = w[(k0 + kb) * C_ + f0 + m]; bb.y = w[(k0 + kb + 1) * C_ + f0 + m];
    acc = wmma_f32(a, bb, acc);
  }

  const int half = lane >> 4;
  const float bcol = bias[f0 + m];
#pragma unroll
  for (int r = 0; r < 8; ++r) {
    const int row = row0 + r + 8 * half;
    y[(size_t)row * C_ + f0 + m] = acc[r] + bcol;
  }
}

// ---------------------------------------------------------------------------
extern "C" void kernel_launch(void* const* d_in, const int* in_sizes, int n_in,
                              void* d_out, int out_size, void* d_ws, size_t ws_size,
                              hipStream_t stream) {
  const float* x     = (const float*)d_in[0];   // [4,64,64,128]
  const float* w_qkv = (const float*)d_in[1];   // [128,384]
  const float* w_out = (const float*)d_in[2];   // [128,128]
  const float* b_out = (const float*)d_in[3];   // [128]
  float* out = (float*)d_out;                   // [4,64,64,128]

  const size_t NEL = (size_t)B_ * NH * S_ * DH; // 2,097,152 floats (8 MB)
  float* qT = (float*)d_ws;                     // [b][h][d][s]
  float* kT = qT + NEL;                         // [b][h][d][s]
  float* vv = kT + NEL;                         // [b][h][s][d]
  float* oo = vv + NEL;                         // [b][s][dim]
  // total workspace use: 32 MB

  qkv_gemm    <<<6144, 128, 0, stream>>>(x, w_qkv, qT, kT, vv);
  l2norm_scale<<<1024, 256, 0, stream>>>(qT, kT);
  flash_attn  <<<512, 256, 0, stream>>>(qT, kT, vv, oo);
  out_gemm    <<<2048, 128, 0, stream>>>(oo, w_out, b_out, out);
}